// DecorrelationPatch2d_53764400611505
// MI455X (gfx1250) — compile-verified
//
#include <hip/hip_runtime.h>
#include <hip/hip_bf16.h>

typedef __attribute__((ext_vector_type(2))) float v2f;
typedef __attribute__((ext_vector_type(8))) float v8f;

#define KH 3
#define KW 3

// ---------------------------------------------------------------------------
// Kernel A: s[m] = sum_k R[m][k]  computed as  S = R * ones  on the matrix
// pipe via V_WMMA_F32_16X16X4_F32. One wave32 per 16-row tile of R.
//
// A-matrix (16x4 f32) ISA layout: lane L holds row M = L&15;
//   VGPR0 = K = (L>>4)*2, VGPR1 = K = (L>>4)*2 + 1.
// B-matrix = all ones, so layout is irrelevant (every element 1.0f).
// C/D (16x16 f32): lanes 0-15 -> N=lane, M=vgpr; lanes 16-31 -> N=lane-16,
//   M=vgpr+8. All columns are identical row sums; lanes with N==0 store them.
// EXEC is all-1s: block == exactly one full wave, no divergence before WMMA.
// ---------------------------------------------------------------------------
__global__ __launch_bounds__(32) void rowsum_wmma_kernel(
    const float* __restrict__ R, float* __restrict__ s, int P) {
  const int lane  = threadIdx.x;          // 0..31
  const int tile  = blockIdx.x;           // 16 rows per tile
  const int m     = lane & 15;
  const int row   = tile * 16 + m;
  const int kbase = (lane >> 4) * 2;      // 0 or 2

  v8f acc = {};
  v2f b;
  b.x = 1.0f;
  b.y = 1.0f;

  const float* rp = R + (size_t)row * (size_t)P + kbase;
  for (int k0 = 0; k0 < P; k0 += 4) {
    v2f a;
    a.x = rp[k0];
    a.y = rp[k0 + 1];
    acc = __builtin_amdgcn_wmma_f32_16x16x4_f32(
        /*neg_a=*/false, a, /*neg_b=*/false, b,
        /*c_mod=*/(short)0, acc, /*reuse_a=*/false, /*reuse_b=*/false);
  }

  if ((lane & 15) == 0) {                 // lanes 0 and 16 -> column N = 0
    const int mbase = (lane >> 4) * 8;    // lane0: M=0..7, lane16: M=8..15
    #pragma unroll
    for (int v = 0; v < 8; ++v) s[tile * 16 + mbase + v] = acc[v];
  }
}

// ---------------------------------------------------------------------------
// Kernel B: out[n,c,h,w] = x[n,c,h,w] * T[c][hclass(h)][wclass(w)]
// with T = (sum of s over the valid 3x3 window) / coverage.
// One block per (n,c) plane; pure b128 streaming at HBM rate.
// ---------------------------------------------------------------------------
__device__ __forceinline__ int border_class(int p, int L) {
  // classes: 0:{0} 1:{0,1} 2:{0,1,2} (interior) 3:{1,2} 4:{2}
  return (p < 2) ? p : ((p > L - 3) ? (p - (L - 5)) : 2);
}

__global__ __launch_bounds__(256) void apply_kernel(
    const float* __restrict__ x, const float* __restrict__ s,
    float* __restrict__ out, int C, int H, int W) {
  __shared__ float sS[KH * KW];
  __shared__ float sM[25];

  const int plane = blockIdx.x;           // n*C + c
  const int c     = plane % C;
  const int t     = threadIdx.x;

  if (t < KH * KW) sS[t] = s[c * KH * KW + t];
  __syncthreads();

  if (t < 25) {
    const int hcl = t / 5, wcl = t % 5;
    const int ilo = (hcl <= 2) ? 0 : hcl - 2;
    const int ihi = (hcl >= 2) ? 2 : hcl;
    const int jlo = (wcl <= 2) ? 0 : wcl - 2;
    const int jhi = (wcl >= 2) ? 2 : wcl;
    float sum = 0.0f;
    for (int i = ilo; i <= ihi; ++i)
      for (int j = jlo; j <= jhi; ++j)
        sum += sS[i * KW + j];
    const float div = (float)((ihi - ilo + 1) * (jhi - jlo + 1));
    sM[t] = sum / div;
  }
  __syncthreads();

  const size_t base = (size_t)plane * (size_t)(H * W);
  const int HW     = H * W;
  const int stride = blockDim.x * 4;

  for (int idx = t * 4; idx < HW; idx += stride) {
    const float4 xv = *(const float4*)(x + base + idx);
    const int h  = idx / W;
    const int w  = idx - h * W;           // chunk stays within one row (W%4==0)
    const int r  = border_class(h, H) * 5;
    float4 o;
    o.x = xv.x * sM[r + border_class(w + 0, W)];
    o.y = xv.y * sM[r + border_class(w + 1, W)];
    o.z = xv.z * sM[r + border_class(w + 2, W)];
    o.w = xv.w * sM[r + border_class(w + 3, W)];
    *(float4*)(out + base + idx) = o;
  }
}

// ---------------------------------------------------------------------------
extern "C" void kernel_launch(void* const* d_in, const int* in_sizes, int n_in,
                              void* d_out, int out_size, void* d_ws, size_t ws_size,
                              hipStream_t stream) {
  const float* x = (const float*)d_in[0];   // [N, C, H, W] f32
  const float* R = (const float*)d_in[1];   // [P, P] f32, P = C*KH*KW
  float* out = (float*)d_out;
  float* s   = (float*)d_ws;                // P floats of scratch

  const int H = 128, W = 128;
  // P = sqrt(in_sizes[1]) (exact; P*P elements in R)
  int P = 1;
  while ((long long)P * (long long)P < (long long)in_sizes[1]) ++P;
  const int C = P / (KH * KW);
  const int N = in_sizes[0] / (C * H * W);

  // s = R * ones via WMMA (P is a multiple of 16)
  rowsum_wmma_kernel<<<P / 16, 32, 0, stream>>>(R, s, P);

  // streaming elementwise apply, one block per (n, c) plane
  apply_kernel<<<N * C, 256, 0, stream>>>(x, s, out, C, H, W);
}